// MTH_diff_line_18622978195631
// MI455X (gfx1250) — compile-verified
//
#include <hip/hip_runtime.h>

// Multi-threshold spiking-neuron scan (MTH), MI455X / gfx1250.
//
//   T = 16 steps, slab = B*C*H*W = 2,097,152 f32 (8 MB); in 142.6 MB, out 142.6 MB
//   -> HBM-bound floor ~12.2 us @ 23.3 TB/s; VALU tuned to stay under that.
//
// Streaming plan: 1 thread = 4 consecutive elements (b128, coalesced, TH_NT —
// single-touch data > 192 MB L2). All 17 loads issued up front (addresses are
// data-independent -> max MLP via split LOADcnt); carry in VGPRs; 17 NT stores.
//
// Bit-exact vs the JAX reference:
//  * mem update keeps left-to-right association ((((mem+xi)-bias)+ei)-eo).
//  * threshold boundaries are T0*2^-k with T0 = RN(0.75*thresh); for
//    non-negative floats, (|m| >= T0*2^-k) == (bits(|m|) >= bits(T0) - k<<23)
//    (integer ordering of non-negative IEEE floats; boundaries all normal).
//    First-fired index: k* = max(0, (bits(T0) - bits(|m|) + 0x7FFFFF) >> 23),
//    fire iff k* <= 7. Exactly reproduces the reference compares.
//  * spike = thresh * (sign(m)*2^-k)  — exact (v_ldexp + exact pow2 scale).
//  * spike/t = RN(thresh/t) * (sign(m)*2^-k) == RN(spike/t) — exact, kills one
//    IEEE divide per step (pow2 scaling commutes with round-to-nearest).
//  * (xi - bias)/t keeps a true IEEE divide (t = 1,2,4,8,16 fold exactly).

typedef float __attribute__((ext_vector_type(4))) v4f;

#define MTH_T   16
#define NUM_TH  8

__global__ __launch_bounds__(256) void mth_scan_kernel(
    const float* __restrict__ x,
    const float* __restrict__ thresh_p,
    float* __restrict__ out,
    int n4) // vec4 columns per timestep slab
{
    const int i = blockIdx.x * blockDim.x + threadIdx.x;
    if (i >= n4) return;

    const float thresh = thresh_p[0];                 // uniform (s_load)
    const float T0     = 0.75f * thresh;              // RN(0.75*thresh)
    const int   T0bits = (int)__builtin_bit_cast(unsigned, T0);   // uniform SGPR

    const v4f* xv = (const v4f*)x;
    v4f*       ov = (v4f*)out;

    // bias = x[0]; preload all T timesteps (data-independent addresses).
    v4f bias = __builtin_nontemporal_load(&xv[i]);
    v4f xs[MTH_T];
    #pragma unroll
    for (int t = 0; t < MTH_T; ++t) {
        const v4f* xt = xv + (size_t)(t + 1) * (size_t)n4;
        xs[t] = __builtin_nontemporal_load(&xt[i]);
    }

    // out[0, ...] = 0
    v4f zero = {0.0f, 0.0f, 0.0f, 0.0f};
    __builtin_nontemporal_store(zero, &ov[i]);

    v4f mem = zero, ei = bias, eo = zero;

    #pragma unroll
    for (int t = 1; t <= MTH_T; ++t) {
        const float tf  = (float)t;
        const float s_t = thresh / tf;                // uniform RN(thresh/t)
        const v4f   xi  = xs[t - 1];
        v4f spike;

        #pragma unroll
        for (int c = 0; c < 4; ++c) {
            // Reference order: mem + xi - bias + exp_in - exp_out
            float m  = ((((mem[c] + xi[c]) - bias[c]) + ei[c]) - eo[c]);

            // --- exact integer quantizer ---
            unsigned mb = __builtin_bit_cast(unsigned, m) & 0x7fffffffu; // bits(|m|)
            int  D    = T0bits - (int)mb;
            int  k    = (D + 0x7fffff) >> 23;         // ceil(D / 2^23)
            int  kc   = k > 0 ? k : 0;                // first-fired index (clamped)
            bool fire = (k <= (NUM_TH - 1));          // |m| >= T0*2^-7 ?

            float ps  = __builtin_copysignf(1.0f, m); // +-1 (v_bfi)
            float p2s = __builtin_ldexpf(ps, -kc);    // +-2^-k, exact (v_ldexp)
            p2s       = fire ? p2s : 0.0f;

            float sp  = thresh * p2s;                 // exact: +-thresh*2^-k (or 0)
            float w   = s_t    * p2s;                 // exact: RN(spike/t)

            float d   = xi[c] - bias[c];
            mem[c]    = m - sp;
            ei[c]     = ei[c] + d / tf;               // true IEEE divide
            eo[c]     = eo[c] + w;
            spike[c]  = sp;
        }

        v4f* ot = ov + (size_t)t * (size_t)n4;
        __builtin_nontemporal_store(spike, &ot[i]);
    }
}

extern "C" void kernel_launch(void* const* d_in, const int* in_sizes, int n_in,
                              void* d_out, int out_size, void* d_ws, size_t ws_size,
                              hipStream_t stream) {
    const float* x      = (const float*)d_in[0];
    const float* thresh = (const float*)d_in[1];
    float*       out    = (float*)d_out;

    const long long total = (long long)in_sizes[0];
    const int step = (int)(total / (MTH_T + 1));   // 2,097,152 elements / timestep
    const int n4   = step / 4;                      // 524,288 vec4 columns

    const int block = 256;
    const int grid  = (n4 + block - 1) / block;     // 2048 blocks
    hipLaunchKernelGGL(mth_scan_kernel, dim3(grid), dim3(block), 0, stream,
                       x, thresh, out, n4);
}